// MultiHeadedRelAttention_20572893347932
// MI455X (gfx1250) — compile-verified
//
#include <hip/hip_runtime.h>
#include <hip/hip_bf16.h>

// ---------------------------------------------------------------------------
// MultiHeadedRelAttention for MI455X (gfx1250, wave32, WMMA).
// B=4, L=512, D=256, H=8, DH=32. rel_table: 12 x 256 (row 11 == 0 padding).
// Key optimization: b_d = (q+v) . rel_table[dist] is precomputed per bucket
// (P[b,h,q,0..11]) turning a 1GB gather into a 12-entry table lookup.
// All GEMMs use v_wmma_f32_16x16x4_f32 (fp32, matches reference precision).
// ---------------------------------------------------------------------------

typedef __attribute__((ext_vector_type(2))) float v2f;
typedef __attribute__((ext_vector_type(8))) float v8f;

#define Bdim 4
#define Ldim 512
#define Ddim 256
#define Hdim 8
#define DHdim 32
#define NBUCKET 12      // rel_table rows (0..11); row 11 is the zero pad row
#define LDSROW 516      // 512 + 4 pad floats -> conflict-free LDS reads

__device__ __forceinline__ v8f wmma_f32_16x16x4(v2f a, v2f b, v8f c) {
    // 8 args: (neg_a, A, neg_b, B, c_mod, C, reuse_a, reuse_b)
    return __builtin_amdgcn_wmma_f32_16x16x4_f32(false, a, false, b,
                                                 (short)0, c, false, false);
}

// Per-wave 16x16 fp32 output tile of C = A(MxK,lda) @ B(KxN,ldb), K%4==0.
// A-frag: lane (l + 16*half) holds A[m0+l][k0+2*half + {0,1}]
// B-frag: lane (l + 16*half) holds B[k0+2*half + {0,1}][n0+l]
__device__ __forceinline__ v8f gemm_tile_16x16(const float* __restrict__ A,
                                               const float* __restrict__ B,
                                               int lda, int ldb, int K,
                                               int m0, int n0, int lane) {
    const int l = lane & 15, half = lane >> 4;
    const float* arow = A + (long)(m0 + l) * lda;
    v8f acc = {};
    for (int k0 = 0; k0 < K; k0 += 4) {
        v2f a, b;
        a.x = arow[k0 + 2 * half];
        a.y = arow[k0 + 2 * half + 1];
        b.x = B[(long)(k0 + 2 * half) * ldb + n0 + l];
        b.y = B[(long)(k0 + 2 * half + 1) * ldb + n0 + l];
        acc = wmma_f32_16x16x4(a, b, acc);
    }
    return acc;
}

// ---------------------------------------------------------------------------
// Kernel 1: fused QKV projection. blockIdx.y selects {Q, K, V}.
// For Q: writes (q/sqrt(DH))+u and (q/sqrt(DH))+v into (B,H,L,DH) layout.
// ---------------------------------------------------------------------------
__global__ void __launch_bounds__(128) qkv_kernel(
    const float* __restrict__ x,
    const float* __restrict__ Wq, const float* __restrict__ bq,
    const float* __restrict__ Wk, const float* __restrict__ bk,
    const float* __restrict__ Wv, const float* __restrict__ bv,
    const float* __restrict__ u, const float* __restrict__ v,
    float* __restrict__ qu, float* __restrict__ qv,
    float* __restrict__ kmat, float* __restrict__ vmat) {
    const int wave = threadIdx.x >> 5;
    const int lane = threadIdx.x & 31;
    const int tile = blockIdx.x * 4 + wave;          // 2048 tiles (128 M x 16 N)
    const int m0 = (tile >> 4) * 16;
    const int n0 = (tile & 15) * 16;
    const int z = blockIdx.y;
    const float* W    = (z == 0) ? Wq : (z == 1) ? Wk : Wv;
    const float* bias = (z == 0) ? bq : (z == 1) ? bk : bv;

    v8f acc = gemm_tile_16x16(x, W, Ddim, Ddim, Ddim, m0, n0, lane);

    const int l = lane & 15, half = lane >> 4;
    const int n = n0 + l;
    const int h = n >> 5, d = n & 31;
    const float bn = bias[n];
    const float un = u[n], vn = v[n];
    const float scale = 0.17677669529663687f;        // 1/sqrt(32)
#pragma unroll
    for (int j = 0; j < 8; ++j) {
        const int m = m0 + j + 8 * half;             // global row = b*L + q
        const int b = m >> 9, q = m & 511;
        const long idx = (((long)(b * Hdim + h)) * Ldim + q) * DHdim + d;
        const float val = acc[j] + bn;
        if (z == 0) {
            const float s = val * scale;
            qu[idx] = s + un;
            qv[idx] = s + vn;
        } else if (z == 1) {
            kmat[idx] = val;
        } else {
            vmat[idx] = val;
        }
    }
}

// ---------------------------------------------------------------------------
// Kernel 2: P[b,h,q,t] = sum_d qv[b,h,q,d] * rel_table[t][h*32+d], t=0..11.
// One wave per (b,h,q) row; shuffle reductions.
// ---------------------------------------------------------------------------
__global__ void __launch_bounds__(128) ptab_kernel(
    const float* __restrict__ qv, const float* __restrict__ rel_table,
    float* __restrict__ P) {
    const int row = (blockIdx.x * blockDim.x + threadIdx.x) >> 5; // (b*H+h)*L+q
    const int lane = threadIdx.x & 31;
    if (row >= Bdim * Hdim * Ldim) return;
    const int h = (row >> 9) & 7;
    const float qvd = qv[(long)row * DHdim + lane];
    const float* rt = rel_table + h * DHdim + lane;
#pragma unroll
    for (int t = 0; t < NBUCKET; ++t) {
        float p = qvd * rt[(long)t * Ddim];
        for (int off = 16; off > 0; off >>= 1) p += __shfl_xor(p, off, 32);
        if (lane == 0) P[(long)row * NBUCKET + t] = p;
    }
}

// ---------------------------------------------------------------------------
// Kernel 3: fused scores + rel-pos lookup + mask + softmax + attn@V.
// One wave per (b, h, 16-row q tile). 1024 blocks of 32 threads.
// ---------------------------------------------------------------------------
__global__ void __launch_bounds__(32) attn_kernel(
    const float* __restrict__ qu, const float* __restrict__ kmat,
    const float* __restrict__ vmat, const float* __restrict__ P,
    const int* __restrict__ distances, const unsigned char* __restrict__ mask,
    float* __restrict__ attn_out, float* __restrict__ ctx) {
    __shared__ float s[16 * LDSROW];                 // 33 KB

    const int lane = threadIdx.x & 31;
    const int l = lane & 15, half = lane >> 4;
    const int bid = blockIdx.x;                      // 0..1023
    const int qt = bid & 31;                         // q tile index
    const int bh = bid >> 5;                         // b*H + h
    const int b = bh >> 3;
    const int h = bh & 7;
    const int q0 = qt * 16;

    const float* quh = qu + (long)bh * Ldim * DHdim;
    const float* kh  = kmat + (long)bh * Ldim * DHdim;
    const float* vh  = vmat + (long)bh * Ldim * DHdim;

    // Preload A fragments for the whole q tile: 16 rows x K=32.
    v2f afr[8];
    {
        const float* arow = quh + (long)(q0 + l) * DHdim;
#pragma unroll
        for (int kk = 0; kk < 8; ++kk) {
            afr[kk].x = arow[4 * kk + 2 * half];
            afr[kk].y = arow[4 * kk + 2 * half + 1];
        }
    }

    // ---- Phase A: scores = qu.k^T + P[dist], masked; stage rows in LDS ----
    for (int kt = 0; kt < 32; ++kt) {
        v8f acc = {};
        const float* brow = kh + (long)(kt * 16 + l) * DHdim;
#pragma unroll
        for (int kk = 0; kk < 8; ++kk) {
            v2f bfr;
            bfr.x = brow[4 * kk + 2 * half];
            bfr.y = brow[4 * kk + 2 * half + 1];
            acc = wmma_f32_16x16x4(afr[kk], bfr, acc);
        }
        const int kpos = kt * 16 + l;
#pragma unroll
        for (int j = 0; j < 8; ++j) {
            const int qrow = j + 8 * half;           // local row
            const int qidx = q0 + qrow;
            const long off = ((long)b * Ldim + qidx) * Ldim + kpos;
            const int dist = distances[off];
            const float bd = P[((long)bh * Ldim + qidx) * NBUCKET + dist];
            float sc = acc[j] + bd;
            if (mask[off]) sc = -1e18f;
            s[qrow * LDSROW + kpos] = sc;
        }
    }

    // ---- Phase B: in-wave softmax per row; write attn to output ----
    float* attn_base = attn_out + ((long)bh * Ldim + q0) * Ldim;
    for (int r = 0; r < 16; ++r) {
        float* row = s + r * LDSROW;
        float vals[16];
        float m = -3.0e38f;
#pragma unroll
        for (int i = 0; i < 16; ++i) {
            vals[i] = row[lane + 32 * i];
            m = fmaxf(m, vals[i]);
        }
        for (int off = 16; off > 0; off >>= 1) m = fmaxf(m, __shfl_xor(m, off, 32));
        float sum = 0.0f;
#pragma unroll
        for (int i = 0; i < 16; ++i) {
            vals[i] = __expf(vals[i] - m);
            sum += vals[i];
        }
        for (int off = 16; off > 0; off >>= 1) sum += __shfl_xor(sum, off, 32);
        const float inv = 1.0f / sum;
        float* arow = attn_base + (long)r * Ldim;
#pragma unroll
        for (int i = 0; i < 16; ++i) {
            const float p = vals[i] * inv;
            row[lane + 32 * i] = p;                  // keep for phase C
            arow[lane + 32 * i] = p;                 // attn output
        }
    }

    // ---- Phase C: ctx(16x32) = attn(16x512, LDS) @ V(512x32) ----
    v8f c0 = {}, c1 = {};
    for (int k0 = 0; k0 < Ldim; k0 += 4) {
        v2f a;
        a.x = s[l * LDSROW + k0 + 2 * half];
        a.y = s[l * LDSROW + k0 + 2 * half + 1];
        const float* b0 = vh + (long)(k0 + 2 * half) * DHdim;
        const float* b1 = vh + (long)(k0 + 2 * half + 1) * DHdim;
        v2f bb;
        bb.x = b0[l];
        bb.y = b1[l];
        c0 = wmma_f32_16x16x4(a, bb, c0);
        bb.x = b0[16 + l];
        bb.y = b1[16 + l];
        c1 = wmma_f32_16x16x4(a, bb, c1);
    }
#pragma unroll
    for (int j = 0; j < 8; ++j) {
        const int qidx = q0 + j + 8 * half;
        const long base = ((long)b * Ldim + qidx) * Ddim + h * DHdim;
        ctx[base + l]      = c0[j];
        ctx[base + 16 + l] = c1[j];
    }
}

// ---------------------------------------------------------------------------
// Kernel 4: output = ctx @ Wo + bo
// ---------------------------------------------------------------------------
__global__ void __launch_bounds__(128) out_kernel(
    const float* __restrict__ ctx, const float* __restrict__ Wo,
    const float* __restrict__ bo, float* __restrict__ out) {
    const int wave = threadIdx.x >> 5;
    const int lane = threadIdx.x & 31;
    const int tile = blockIdx.x * 4 + wave;
    const int m0 = (tile >> 4) * 16;
    const int n0 = (tile & 15) * 16;
    v8f acc = gemm_tile_16x16(ctx, Wo, Ddim, Ddim, Ddim, m0, n0, lane);
    const int l = lane & 15, half = lane >> 4;
    const float bn = bo[n0 + l];
#pragma unroll
    for (int j = 0; j < 8; ++j)
        out[(long)(m0 + j + 8 * half) * Ddim + n0 + l] = acc[j] + bn;
}

// ---------------------------------------------------------------------------
extern "C" void kernel_launch(void* const* d_in, const int* in_sizes, int n_in,
                              void* d_out, int out_size, void* d_ws, size_t ws_size,
                              hipStream_t stream) {
    const float* x             = (const float*)d_in[0];
    const unsigned char* mask  = (const unsigned char*)d_in[1]; // bool -> 1 byte
    const int* dist            = (const int*)d_in[2];
    const float* Wq = (const float*)d_in[3];
    const float* bq = (const float*)d_in[4];
    const float* Wk = (const float*)d_in[5];
    const float* bk = (const float*)d_in[6];
    const float* Wv = (const float*)d_in[7];
    const float* bv = (const float*)d_in[8];
    const float* Wo = (const float*)d_in[9];
    const float* bo = (const float*)d_in[10];
    const float* rel = (const float*)d_in[11];
    const float* u  = (const float*)d_in[12];
    const float* v  = (const float*)d_in[13];

    float* out  = (float*)d_out;                     // (B, L, D)
    float* attn = out + (long)Bdim * Ldim * Ddim;    // (B, H, L, L)

    // Workspace layout (floats): qu, qv, k, v in (B,H,L,DH); P; ctx in (B,L,D)
    const long NH = (long)Bdim * Hdim * Ldim * DHdim; // 524288
    float* ws  = (float*)d_ws;
    float* qu  = ws;
    float* qv  = qu + NH;
    float* km  = qv + NH;
    float* vm  = km + NH;
    float* P   = vm + NH;                            // 16384 * 12
    float* ctx = P + (long)Bdim * Hdim * Ldim * NBUCKET;

    qkv_kernel<<<dim3(512, 3), 128, 0, stream>>>(x, Wq, bq, Wk, bk, Wv, bv,
                                                 u, v, qu, qv, km, vm);
    ptab_kernel<<<4096, 128, 0, stream>>>(qv, rel, P);
    attn_kernel<<<1024, 32, 0, stream>>>(qu, km, vm, P, dist, mask, attn, ctx);
    out_kernel<<<512, 128, 0, stream>>>(ctx, Wo, bo, out);
}